// GAT_1022202216997
// MI455X (gfx1250) — compile-verified
//
#include <hip/hip_runtime.h>
#include <hip/hip_bf16.h>
#include <math.h>

// ---------------------------------------------------------------------------
// GAT 2-layer forward for MI455X (gfx1250, wave32, WGP mode).
//  * Dense projections: V_WMMA_F32_16X16X4_F32 (exact fp32 matrix pipe),
//    A panel staged in LDS; K/Nn are compile-time so all B/C addresses fold
//    into 24-bit immediate offsets (no per-iteration 64-bit adds).
//  * Edge softmax/aggregation: 3-pass atomic segment reduction; the ~140 MB
//    working set fits in the 192 MB L2, scatter is channel-coalesced.
// ---------------------------------------------------------------------------

#define GAT_N      50000
#define GAT_E      800000
#define GAT_EE     (GAT_E + GAT_N)   // edges + self loops
#define GAT_INCH   128
#define GAT_H1     16                // heads, layer 1
#define GAT_C1     16                // hidden per head, layer 1
#define GAT_F1     (GAT_H1 * GAT_C1) // 256
#define GAT_F2     64                // out channels, layer 2 (1 head)
#define NEG_SLOPE  0.2f
#define GAT_EPS    1e-16f

typedef float v2f __attribute__((ext_vector_type(2)));
typedef float v8f __attribute__((ext_vector_type(8)));

// ---------------------------------------------------------------------------
// fp32 WMMA GEMM: C[M,Nn] = A[M,K] * B[K,Nn], row-major, M%16==0; K,Nn static.
// One 128-thread block (4 waves) per 16-row M-slab:
//   - stage A[16,K] into LDS with coalesced float4 loads
//   - each wave owns N-tiles nt = wave, wave+4, ...; K loop unrolled x8 on
//     V_WMMA_F32_16X16X4_F32 with immediate-offset global loads for B.
// Fragment layouts per CDNA5 ISA 7.12.2:
//   A 16x4 f32 : lanes 0-15 -> M=lane, VGPRs = K{0,1}; lanes 16-31 -> K{2,3}
//   B 4x16  f32: lanes 0-15 -> N=lane, VGPRs = K{0,1}; lanes 16-31 -> K{2,3}
//   C 16x16    : VGPR i, lanes 0-15 -> M=i, lanes 16-31 -> M=i+8; N=lane&15
// ---------------------------------------------------------------------------
#define GEMM_THREADS 128
#define GEMM_WAVES   4

template <int K, int Nn>
__global__ __launch_bounds__(GEMM_THREADS) void gat_wmma_gemm(
    const float* __restrict__ A, const float* __restrict__ B,
    float* __restrict__ C)
{
    __shared__ float ldsA[16 * K];                 // 8-16 KB of 320 KB WGP LDS
    const int tid = threadIdx.x;
    const int m0  = blockIdx.x * 16;

    // --- coalesced cooperative stage of A[m0:m0+16, :] into LDS ---
    {
        const float4* Ag = (const float4*)(A + (size_t)m0 * K);
        float4*       L4 = (float4*)ldsA;
#pragma unroll
        for (int i = 0; i < (16 * K / 4) / GEMM_THREADS; ++i)
            L4[tid + i * GEMM_THREADS] = Ag[tid + i * GEMM_THREADS];
    }
    __syncthreads();

    const int lane = tid & 31;
    const int wave = tid >> 5;
    const int l    = lane & 15;
    const int half = lane >> 4;                    // K pair {0,1} vs {2,3}

    const float* aRow = ldsA + l * K + half * 2;   // ds_load_b64 per k-step

#pragma unroll
    for (int nt = wave; nt < Nn / 16; nt += GEMM_WAVES) {
        const int n0 = nt * 16;
        // base pointer: row (half*2), column n0+l; K-steps become literal offsets
        const float* Bcol = B + (size_t)(half * 2) * Nn + n0 + l;
        v8f acc = {};
#pragma unroll 8
        for (int k = 0; k < K; k += 4) {
            v2f a = *(const v2f*)(aRow + k);       // A[l][k+half*2 .. +1]
            v2f b;
            b.x = Bcol[k * Nn];                    // B[k+half*2  ][n0+l]
            b.y = Bcol[(k + 1) * Nn];              // B[k+half*2+1][n0+l]
            // 8 args: (neg_a, A, neg_b, B, c_mod, C, reuse_a, reuse_b)
            acc = __builtin_amdgcn_wmma_f32_16x16x4_f32(
                false, a, false, b, (short)0, acc, false, false);
        }
        float* Crow = C + (size_t)(m0 + half * 8) * Nn + n0 + l;
#pragma unroll
        for (int i = 0; i < 8; ++i)
            Crow[i * Nn] = acc[i];
    }
}

// ---------------------------------------------------------------------------
// Helpers
// ---------------------------------------------------------------------------
__device__ __forceinline__ void atomicMaxF32(float* addr, float val) {
    // Monotonic int/uint trick (valid for all finite floats).
    if (val >= 0.0f) atomicMax((int*)addr, __float_as_int(val));
    else             atomicMin((unsigned int*)addr, __float_as_uint(val));
}

__device__ __forceinline__ void edge_nodes(int e, const int* __restrict__ ei,
                                           int& s, int& d) {
    if (e < GAT_E) { s = ei[e]; d = ei[GAT_E + e]; }
    else           { s = e - GAT_E; d = s; }      // appended self loop
}

__device__ __forceinline__ float leaky(float v) {
    return v > 0.0f ? v : NEG_SLOPE * v;
}

__global__ void gat_fill(float* __restrict__ p, float v, int n) {
    int i = blockIdx.x * blockDim.x + threadIdx.x;
    if (i < n) p[i] = v;
}

// as[n,H], ad[n,H]: per-node attention logits (dot over C channels).
__global__ void gat_node_alpha(const float* __restrict__ h,
                               const float* __restrict__ a_s,
                               const float* __restrict__ a_d,
                               float* __restrict__ as, float* __restrict__ ad,
                               int nnodes, int H, int C)
{
    int i = blockIdx.x * blockDim.x + threadIdx.x;
    if (i >= nnodes * H) return;
    int hd = i % H;
    const float* hp  = h + (size_t)i * C;
    const float* asv = a_s + (size_t)hd * C;
    const float* adv = a_d + (size_t)hd * C;
    float s1 = 0.0f, s2 = 0.0f;
    for (int c = 0; c < C; ++c) { float v = hp[c]; s1 += v * asv[c]; s2 += v * adv[c]; }
    as[i] = s1; ad[i] = s2;
}

// Pass 1: segment max of leaky-relu logits over destination. (edge, head) grain.
__global__ void gat_edge_max(const int* __restrict__ ei,
                             const float* __restrict__ as,
                             const float* __restrict__ ad,
                             float* __restrict__ m, int H)
{
    int i = blockIdx.x * blockDim.x + threadIdx.x;
    if (i >= GAT_EE * H) return;
    int e = i / H, hd = i - e * H;
    int s, d; edge_nodes(e, ei, s, d);
    float v = leaky(as[s * H + hd] + ad[d * H + hd]);
    atomicMaxF32(&m[d * H + hd], v);
}

// Pass 2: segment sum of exp(e - max). (edge, head) grain.
__global__ void gat_edge_denom(const int* __restrict__ ei,
                               const float* __restrict__ as,
                               const float* __restrict__ ad,
                               const float* __restrict__ m,
                               float* __restrict__ den, int H)
{
    int i = blockIdx.x * blockDim.x + threadIdx.x;
    if (i >= GAT_EE * H) return;
    int e = i / H, hd = i - e * H;
    int s, d; edge_nodes(e, ei, s, d);
    int dH = d * H + hd;
    float v = leaky(as[s * H + hd] + ad[dH]);
    atomicAdd(&den[dH], expf(v - m[dH]));
}

// Pass 3: weighted message scatter at (edge, channel) grain so that both the
// h[src] gather and the atomicAdd to agg[dst] are contiguous across a wave.
// F = H*C total feature width. Softmax scalars are L2-hot; recompute is free.
__global__ void gat_edge_msg(const int* __restrict__ ei,
                             const float* __restrict__ as,
                             const float* __restrict__ ad,
                             const float* __restrict__ m,
                             const float* __restrict__ den,
                             const float* __restrict__ h,
                             float* __restrict__ agg, int H, int C, int F)
{
    int i = blockIdx.x * blockDim.x + threadIdx.x;   // < EE*F  (max 217.6M)
    if (i >= GAT_EE * F) return;
    int e  = i / F;
    int j  = i - e * F;                              // channel within F
    int hd = j / C;
    int s, d; edge_nodes(e, ei, s, d);
    int dH = d * H + hd;
    float v     = leaky(as[s * H + hd] + ad[dH]);
    float alpha = expf(v - m[dH]) / (den[dH] + GAT_EPS);
    atomicAdd(&agg[(size_t)d * F + j], h[(size_t)s * F + j] * alpha);
}

__global__ void gat_bias_add(float* __restrict__ p, const float* __restrict__ b,
                             int total, int F)
{
    int i = blockIdx.x * blockDim.x + threadIdx.x;
    if (i < total) p[i] += b[i % F];
}

__global__ void gat_finalize(const float* __restrict__ agg,
                             const float* __restrict__ b,
                             float* __restrict__ out, int total, int F)
{
    int i = blockIdx.x * blockDim.x + threadIdx.x;
    if (i >= total) return;
    float v = agg[i] + b[i % F];
    out[i] = 1.0f / (1.0f + expf(-v));
}

// ---------------------------------------------------------------------------
// Host launcher
// ---------------------------------------------------------------------------
extern "C" void kernel_launch(void* const* d_in, const int* in_sizes, int n_in,
                              void* d_out, int out_size, void* d_ws, size_t ws_size,
                              hipStream_t stream)
{
    (void)in_sizes; (void)n_in; (void)out_size; (void)ws_size;

    const float* x      = (const float*)d_in[0];
    const int*   ei     = (const int*)  d_in[1];   // [2,E] row-major int32
    const float* W1     = (const float*)d_in[2];   // [128,256]
    const float* a_src1 = (const float*)d_in[3];   // [16,16]
    const float* a_dst1 = (const float*)d_in[4];
    const float* b1     = (const float*)d_in[5];   // [256]
    const float* W2     = (const float*)d_in[6];   // [256,64]
    const float* a_src2 = (const float*)d_in[7];   // [1,64]
    const float* a_dst2 = (const float*)d_in[8];
    const float* b2     = (const float*)d_in[9];   // [64]
    float*       out    = (float*)d_out;           // [N,64]

    // Workspace layout (floats). Total ~35.4M floats = ~142 MB (fits in L2).
    float* ws   = (float*)d_ws;
    float* h1   = ws;                                // N*256
    float* as1  = h1   + (size_t)GAT_N * GAT_F1;     // N*16
    float* ad1  = as1  + (size_t)GAT_N * GAT_H1;     // N*16
    float* m1   = ad1  + (size_t)GAT_N * GAT_H1;     // N*16
    float* den1 = m1   + (size_t)GAT_N * GAT_H1;     // N*16
    float* agg1 = den1 + (size_t)GAT_N * GAT_H1;     // N*256
    float* h2   = agg1 + (size_t)GAT_N * GAT_F1;     // N*64
    float* as2  = h2   + (size_t)GAT_N * GAT_F2;     // N
    float* ad2  = as2  + (size_t)GAT_N;              // N
    float* m2   = ad2  + (size_t)GAT_N;              // N
    float* den2 = m2   + (size_t)GAT_N;              // N
    float* agg2 = den2 + (size_t)GAT_N;              // N*64

    const int TB = 256;
    auto blocks = [](long long n, int tb) { return (int)((n + tb - 1) / tb); };

    // ---------------- Layer 1 ----------------
    // h1 = x @ W1   (50000x128 @ 128x256); LDS stage = 8 KB
    gat_wmma_gemm<GAT_INCH, GAT_F1><<<GAT_N / 16, GEMM_THREADS, 0, stream>>>(
        x, W1, h1);

    gat_node_alpha<<<blocks((long long)GAT_N * GAT_H1, TB), TB, 0, stream>>>(
        h1, a_src1, a_dst1, as1, ad1, GAT_N, GAT_H1, GAT_C1);

    gat_fill<<<blocks((long long)GAT_N * GAT_H1, TB), TB, 0, stream>>>(m1,  -1e30f, GAT_N * GAT_H1);
    gat_fill<<<blocks((long long)GAT_N * GAT_H1, TB), TB, 0, stream>>>(den1, 0.0f,  GAT_N * GAT_H1);
    gat_fill<<<blocks((long long)GAT_N * GAT_F1, TB), TB, 0, stream>>>(agg1, 0.0f,  GAT_N * GAT_F1);

    gat_edge_max  <<<blocks((long long)GAT_EE * GAT_H1, TB), TB, 0, stream>>>(
        ei, as1, ad1, m1, GAT_H1);
    gat_edge_denom<<<blocks((long long)GAT_EE * GAT_H1, TB), TB, 0, stream>>>(
        ei, as1, ad1, m1, den1, GAT_H1);
    gat_edge_msg  <<<blocks((long long)GAT_EE * GAT_F1, TB), TB, 0, stream>>>(
        ei, as1, ad1, m1, den1, h1, agg1, GAT_H1, GAT_C1, GAT_F1);

    // agg1 += b1  -> layer-2 input
    gat_bias_add<<<blocks((long long)GAT_N * GAT_F1, TB), TB, 0, stream>>>(
        agg1, b1, GAT_N * GAT_F1, GAT_F1);

    // ---------------- Layer 2 (1 head, 64 ch, mean == identity) ----------------
    // h2 = agg1 @ W2   (50000x256 @ 256x64); LDS stage = 16 KB
    gat_wmma_gemm<GAT_F1, GAT_F2><<<GAT_N / 16, GEMM_THREADS, 0, stream>>>(
        agg1, W2, h2);

    gat_node_alpha<<<blocks((long long)GAT_N, TB), TB, 0, stream>>>(
        h2, a_src2, a_dst2, as2, ad2, GAT_N, 1, GAT_F2);

    gat_fill<<<blocks((long long)GAT_N, TB), TB, 0, stream>>>(m2,  -1e30f, GAT_N);
    gat_fill<<<blocks((long long)GAT_N, TB), TB, 0, stream>>>(den2, 0.0f,  GAT_N);
    gat_fill<<<blocks((long long)GAT_N * GAT_F2, TB), TB, 0, stream>>>(agg2, 0.0f, GAT_N * GAT_F2);

    gat_edge_max  <<<blocks((long long)GAT_EE, TB), TB, 0, stream>>>(
        ei, as2, ad2, m2, 1);
    gat_edge_denom<<<blocks((long long)GAT_EE, TB), TB, 0, stream>>>(
        ei, as2, ad2, m2, den2, 1);
    gat_edge_msg  <<<blocks((long long)GAT_EE * GAT_F2, TB), TB, 0, stream>>>(
        ei, as2, ad2, m2, den2, h2, agg2, 1, GAT_F2, GAT_F2);

    // out = sigmoid(agg2 + b2)
    gat_finalize<<<blocks((long long)GAT_N * GAT_F2, TB), TB, 0, stream>>>(
        agg2, b2, out, GAT_N * GAT_F2, GAT_F2);
}